// UMGMQuantizer_39092792328256
// MI455X (gfx1250) — compile-verified
//
#include <hip/hip_runtime.h>
#include <hip/hip_bf16.h>
#include <math.h>

// ---------------------------------------------------------------------------
// UMGMQuantizer forward for MI455X (gfx1250, wave32, WMMA).
// Forward-pass algebra: straight-through quant == one-hot(argmax(logit+g)),
// so grouped K->C convs are gathers; softmax cancels. All GEMMs run on
// V_WMMA_F32_16X16X4_F32 (native fp32 matrix op). Gumbel argmax is fused into
// the quantizer GEMM via packed u64 atomic-max, so the 176 MB logit tensor is
// write-only (non-temporal stores keep it out of the 192 MB L2).
// ---------------------------------------------------------------------------

typedef __attribute__((ext_vector_type(2))) float v2f;
typedef __attribute__((ext_vector_type(4))) float v4f;
typedef __attribute__((ext_vector_type(8))) float v8f;

__device__ __forceinline__ v8f wmma_f32_k4(v2f a, v2f b, v8f c) {
  return __builtin_amdgcn_wmma_f32_16x16x4_f32(
      false, a, false, b, (short)0, c, false, false);
}

#define CCH 256
#define NK  4096
#define NB  4

// ---- optional gfx1250 async global->LDS staging --------------------------
#define AS1 __attribute__((address_space(1)))
#define AS3 __attribute__((address_space(3)))
#if defined(__has_builtin)
#if __has_builtin(__builtin_amdgcn_global_load_async_to_lds_b32) && \
    __has_builtin(__builtin_amdgcn_s_wait_asynccnt)
#define USE_ASYNC_LDS 1
#endif
#endif
#ifndef USE_ASYNC_LDS
#define USE_ASYNC_LDS 0
#endif

// Stage a [IC x 16] activation tile (element (ic,n) = src[ic*HW+n]) into LDS,
// float2-packed as bsf[(ic>>1)*32 + n*2 + (ic&1)] so the compute loop reads
// one aligned b64 per WMMA k-step with conflict-free banking.
__device__ __forceinline__ void stage_tile_f2(const float* __restrict__ src,
                                              int HW, int IC,
                                              float* __restrict__ bsf,
                                              int lane) {
#if USE_ASYNC_LDS
  AS3 int* lb = (AS3 int*)(unsigned)(size_t)bsf;
  for (int idx = lane; idx < IC * 16; idx += 32) {
    const int ic = idx >> 4, n2 = idx & 15;
    AS1 int* gp = (AS1 int*)(size_t)(const void*)(src + (size_t)ic * HW + n2);
    AS3 int* lp = lb + ((ic >> 1) * 32 + n2 * 2 + (ic & 1));
    __builtin_amdgcn_global_load_async_to_lds_b32(gp, lp, 0, 0);
  }
  __builtin_amdgcn_s_wait_asynccnt(0);
  __syncthreads();
#else
  for (int idx = lane; idx < IC * 16; idx += 32) {
    const int ic = idx >> 4, n2 = idx & 15;
    bsf[(ic >> 1) * 32 + n2 * 2 + (ic & 1)] = src[(size_t)ic * HW + n2];
  }
  __syncthreads();
#endif
}

// ---------------------------------------------------------------------------
// 3x3 stride-2 pad-1 conv 256->256, implicit GEMM. Boundary handling is
// branchless: loads use clamped addresses, the pad-zero is a v_cndmask.
// grid = (HWo/16, 16, B), block = 32 (one wave per 16x16 D tile).
// ---------------------------------------------------------------------------
__global__ void conv3x3s2_wmma(const float* __restrict__ in,
                               const float* __restrict__ wgt,
                               float* __restrict__ out, int Hin, int Hout) {
  const int lane = threadIdx.x;
  const int half = lane >> 4, nn = lane & 15;
  const int n0 = blockIdx.x * 16;
  const int oc0 = blockIdx.y * 16;
  const int b = blockIdx.z;
  const int HWo = Hout * Hout;
  const size_t HH = (size_t)Hin * Hin;
  const float* inb = in + (size_t)b * CCH * HH;

  const int n = n0 + nn;
  const int oy = n / Hout, ox = n % Hout;

  v8f acc = {};
  for (int ky = 0; ky < 3; ++ky) {
    const int iy = 2 * oy + ky - 1;
    for (int kx = 0; kx < 3; ++kx) {
      const int ix = 2 * ox + kx - 1;
      const bool ok = (iy >= 0) & (iy < Hin) & (ix >= 0) & (ix < Hin);
      const int iyc = min(max(iy, 0), Hin - 1);
      const int ixc = min(max(ix, 0), Hin - 1);
      const float* ip = inb + (size_t)iyc * Hin + ixc;   // always valid
      const float* wp = wgt + (size_t)(oc0 + nn) * CCH * 9 + ky * 3 + kx;
      for (int c0 = 0; c0 < CCH; c0 += 4) {
        v2f a, bb;
        a.x = wp[(size_t)(c0 + 2 * half) * 9];
        a.y = wp[(size_t)(c0 + 2 * half + 1) * 9];
        const float x0 = ip[(size_t)(c0 + 2 * half) * HH];
        const float x1 = ip[(size_t)(c0 + 2 * half + 1) * HH];
        bb.x = ok ? x0 : 0.0f;   // v_cndmask, no EXEC branching
        bb.y = ok ? x1 : 0.0f;
        acc = wmma_f32_k4(a, bb, acc);
      }
    }
  }
  float* ob = out + (size_t)b * CCH * HWo;
  for (int r = 0; r < 8; ++r)
    ob[(size_t)(oc0 + r + 8 * half) * HWo + n0 + nn] = acc[r];
}

// ---------------------------------------------------------------------------
// Dense 1x1 conv as GEMM with LDS-staged B-tile, sweeping all OC/16 output
// tiles (16x activation reuse). grid = (HW/16, 1, B), block = 32.
// ---------------------------------------------------------------------------
__global__ void gemm1x1_wmma(const float* __restrict__ in,
                             const float* __restrict__ wm,
                             float* __restrict__ out,
                             int IC, int OC, int HW, int accumulate) {
  __shared__ float bsf[(CCH / 2) * 32];  // 16 KB max
  const int lane = threadIdx.x;
  const int half = lane >> 4, nn = lane & 15;
  const int n0 = blockIdx.x * 16;
  const int b = blockIdx.z;

  stage_tile_f2(in + (size_t)b * IC * HW + n0, HW, IC, bsf, lane);

  float* ob = out + (size_t)b * OC * HW;
  const int nOT = OC / 16;
  for (int ot = 0; ot < nOT; ++ot) {
    const int oc0 = ot * 16;
    const v2f* w2 = (const v2f*)(wm + (size_t)(oc0 + nn) * IC);
    if (ot + 1 < nOT)
      __builtin_prefetch(wm + (size_t)(oc0 + 16 + nn) * IC, 0, 0);
    v8f acc = {};
    for (int c0 = 0; c0 < IC; c0 += 4) {
      const v2f a = w2[(c0 >> 1) + half];
      const v2f bb = *(const v2f*)&bsf[((c0 >> 1) + half) * 32 + nn * 2];
      acc = wmma_f32_k4(a, bb, acc);
    }
    for (int r = 0; r < 8; ++r) {
      const size_t oi = (size_t)(oc0 + r + 8 * half) * HW + n0 + nn;
      float v = acc[r];
      if (accumulate) v += ob[oi];
      ob[oi] = v;
    }
  }
}

// ---------------------------------------------------------------------------
// Grouped quantizer projection + fused Gumbel argmax.
// logit[b, m*K+k, p] = sum_{c<128} wq[m*K+k, c] * zh[b, m*128+c, p]
// Logits leave via non-temporal stores (write-only output). Per 16-code tile
// each lane reduces its 8 (k,p) values with Gumbel noise, combines with its
// cross-half partner (same p), and atomically max-merges a monotonic
// (floatkey<<32 | ~k) packing -> ties pick the smallest k, like jnp.argmax.
// grid = (HW/16, 16, B*M), block = 32.
// ---------------------------------------------------------------------------
__global__ void quantize_logits_wmma(const float* __restrict__ zh,
                                     const float* __restrict__ wq,
                                     const float* __restrict__ u,
                                     float* __restrict__ logit,
                                     unsigned long long* __restrict__ pack,
                                     int HW) {
  __shared__ float bsf[64 * 32];  // 8 KB (128 ic)
  const int lane = threadIdx.x;
  const int half = lane >> 4, nn = lane & 15;
  const int n0 = blockIdx.x * 16;
  const int bm = blockIdx.z;           // == b*2 + m
  const int m = bm & 1, b = bm >> 1;

  stage_tile_f2(zh + (size_t)(b * CCH + m * 128) * HW + n0, HW, 128, bsf, lane);

  const float* wg = wq + (size_t)m * NK * 128;
  float* ob = logit + (size_t)bm * NK * HW;
  const float* up = u + ((size_t)bm * HW + n0 + nn) * NK;
  unsigned long long* cell = pack + (size_t)bm * HW + n0 + nn;

  for (int kt = 0; kt < 16; ++kt) {
    const int k0 = blockIdx.y * 256 + kt * 16;
    const v2f* w2 = (const v2f*)(wg + (size_t)(k0 + nn) * 128);
    if (kt + 1 < 16)
      __builtin_prefetch(wg + (size_t)(k0 + 16 + nn) * 128, 0, 0);
    v8f acc = {};
    for (int c0 = 0; c0 < 128; c0 += 4) {
      const v2f a = w2[(c0 >> 1) + half];
      const v2f bb = *(const v2f*)&bsf[((c0 >> 1) + half) * 32 + nn * 2];
      acc = wmma_f32_k4(a, bb, acc);
    }
    const int kbase = k0 + 8 * half;
    const v4f u0v = __builtin_nontemporal_load((const v4f*)&up[kbase]);
    const v4f u1v = __builtin_nontemporal_load((const v4f*)&up[kbase + 4]);
    float bestv = -INFINITY;
    int bestk = 0;
    for (int r = 0; r < 8; ++r) {
      const int k = kbase + r;
      __builtin_nontemporal_store(acc[r], &ob[(size_t)k * HW + n0 + nn]);
      const float uu = (r < 4) ? u0v[r] : u1v[r - 4];
      const float v = acc[r] - __logf(-__logf(uu));  // logit + gumbel
      if (v > bestv) { bestv = v; bestk = k; }
    }
    const float ov = __shfl_xor(bestv, 16, 32);
    const int ok = __shfl_xor(bestk, 16, 32);
    if (ov > bestv || (ov == bestv && ok < bestk)) { bestv = ov; bestk = ok; }
    if (half == 0) {
      unsigned kb = __float_as_uint(bestv);
      kb = (kb & 0x80000000u) ? ~kb : (kb | 0x80000000u);  // monotonic key
      const unsigned long long pk =
          ((unsigned long long)kb << 32) | (unsigned)(~bestk);
      atomicMax(cell, pk);
    }
  }
}

__global__ void zero_pack_kernel(unsigned long long* __restrict__ pack) {
  pack[(size_t)blockIdx.x * 256 + threadIdx.x] = 0ull;
}

__global__ void decode_code_kernel(const unsigned long long* __restrict__ pack,
                                   int* __restrict__ code_i,
                                   float* __restrict__ code_f) {
  const size_t i = (size_t)blockIdx.x * 256 + threadIdx.x;
  const unsigned k = ~(unsigned)pack[i];
  code_i[i] = (int)k;
  code_f[i] = (float)k;
}

// ---------------------------------------------------------------------------
//   lh != null : out = lh - w_dq[ch, code]   (encoder residual)
//   lh == null : out = w_dq[ch, code]        (decoder dequant of one-hot)
// ---------------------------------------------------------------------------
__global__ void gather_dq_kernel(const float* __restrict__ lh,
                                 const float* __restrict__ wdq,
                                 const int* __restrict__ code,
                                 float* __restrict__ out, int HW) {
  const size_t idx = (size_t)blockIdx.x * blockDim.x + threadIdx.x;
  const int p = (int)(idx % HW);
  const size_t t = idx / HW;
  const int ch = (int)(t % CCH);
  const int b = (int)(t / CCH);
  const int m = ch >> 7;
  const int k = code[(size_t)(b * 2 + m) * HW + p];
  const float gv = wdq[(size_t)ch * NK + k];
  out[idx] = lh ? (lh[idx] - gv) : gv;
}

// ---------------------------------------------------------------------------
// 2x2 stride-2 conv-transpose: out[b,o,2y+r,2x+s] = sum_c t[b,c,y,x] *
// w_rest[o,c,1-r,1-s]. One LDS staging feeds all 4 parities x 16 oc tiles.
// grid = (HWin/16, 1, B), block = 32.
// ---------------------------------------------------------------------------
__global__ void convT2x2_wmma(const float* __restrict__ tin,
                              const float* __restrict__ wrest,
                              float* __restrict__ out, int Hin) {
  __shared__ float bsf[(CCH / 2) * 32];  // 16 KB
  const int lane = threadIdx.x;
  const int half = lane >> 4, nn = lane & 15;
  const int n0 = blockIdx.x * 16;
  const int b = blockIdx.z;
  const int HW = Hin * Hin, Hout = 2 * Hin;

  stage_tile_f2(tin + (size_t)b * CCH * HW + n0, HW, CCH, bsf, lane);

  const int n = n0 + nn;
  const int y = n / Hin, x = n % Hin;
  float* ob = out + (size_t)b * CCH * Hout * Hout;

  for (int rs = 0; rs < 4; ++rs) {
    const int r = rs >> 1, s = rs & 1;
    const int kyx = (1 - r) * 2 + (1 - s);
    const size_t pos = (size_t)(2 * y + r) * Hout + (2 * x + s);
    for (int ot = 0; ot < 16; ++ot) {
      const int oc0 = ot * 16;
      const float* wrow = wrest + (size_t)(oc0 + nn) * CCH * 4 + kyx;
      v8f acc = {};
      for (int c0 = 0; c0 < CCH; c0 += 4) {
        v2f a;
        a.x = wrow[(size_t)(c0 + 2 * half) * 4];
        a.y = wrow[(size_t)(c0 + 2 * half + 1) * 4];
        const v2f bb = *(const v2f*)&bsf[((c0 >> 1) + half) * 32 + nn * 2];
        acc = wmma_f32_k4(a, bb, acc);
      }
      for (int rr = 0; rr < 8; ++rr)
        ob[(size_t)(oc0 + rr + 8 * half) * Hout * Hout + pos] = acc[rr];
    }
  }
}

// ---------------------------------------------------------------------------
extern "C" void kernel_launch(void* const* d_in, const int* in_sizes, int n_in,
                              void* d_out, int out_size, void* d_ws,
                              size_t ws_size, hipStream_t stream) {
  (void)in_sizes; (void)n_in; (void)out_size; (void)ws_size;

  const float* x = (const float*)d_in[0];
  const float* u[3] = {(const float*)d_in[1], (const float*)d_in[2],
                       (const float*)d_in[3]};
  const float *w_enc[3], *w_qh[3], *w_q[3], *w_dq[3], *w_dh[3], *w_rest[3];
  for (int i = 0; i < 3; ++i) {
    w_enc[i]  = (const float*)d_in[4 + 6 * i];
    w_qh[i]   = (const float*)d_in[5 + 6 * i];
    w_q[i]    = (const float*)d_in[6 + 6 * i];
    w_dq[i]   = (const float*)d_in[7 + 6 * i];
    w_dh[i]   = (const float*)d_in[8 + 6 * i];
    w_rest[i] = (const float*)d_in[9 + 6 * i];
  }
  const float* w_lh[2]   = {(const float*)d_in[22], (const float*)d_in[23]};
  const float* w_side[2] = {(const float*)d_in[24], (const float*)d_in[25]};

  const int hs[3] = {32, 16, 8};

  // ---- d_out layout: former | code0..2 | logit0..2 ----
  float* out = (float*)d_out;
  float* out_former = out;
  size_t off = (size_t)NB * CCH * 64 * 64;
  float* out_code[3];
  for (int i = 0; i < 3; ++i) { out_code[i] = out + off; off += (size_t)NB * 2 * hs[i] * hs[i]; }
  float* out_logit[3];
  for (int i = 0; i < 3; ++i) { out_logit[i] = out + off; off += (size_t)NB * 2 * NK * hs[i] * hs[i]; }

  // ---- workspace ----
  float* wsf = (float*)d_ws;
  size_t o = 0;
  const size_t A32 = (size_t)NB * CCH * 32 * 32;  // 1 Mi floats
  const size_t A16 = (size_t)NB * CCH * 16 * 16;
  float* z    = wsf + o; o += A32;
  float* zh   = wsf + o; o += A32;   // reused as decoder t-buffer
  float* lh   = wsf + o; o += A32;   // reused as decoder gather buffer
  float* cur1 = wsf + o; o += A32;
  float* cur2 = wsf + o; o += A16;
  float* f16b = wsf + o; o += A16;
  float* f32b = wsf + o; o += A32;
  int* code_i[3];
  code_i[0] = (int*)(wsf + o);
  code_i[1] = code_i[0] + NB * 2 * 32 * 32;
  code_i[2] = code_i[1] + NB * 2 * 16 * 16;
  o += (NB * 2 * (32 * 32 + 16 * 16 + 8 * 8) + 1) & ~1ull;  // ints, even pad
  unsigned long long* packb[3];
  packb[0] = (unsigned long long*)(wsf + o);
  packb[1] = packb[0] + NB * 2 * 32 * 32;
  packb[2] = packb[1] + NB * 2 * 16 * 16;
  const int npack = NB * 2 * (32 * 32 + 16 * 16 + 8 * 8);  // 10752

  zero_pack_kernel<<<dim3(npack / 256), 256, 0, stream>>>(packb[0]);

  // ================= encoder =================
  const float* cur = x;
  int Hin = 64;
  for (int i = 0; i < 3; ++i) {
    const int Ho = hs[i];
    const int HW = Ho * Ho;
    const int ncell = NB * 2 * HW;
    conv3x3s2_wmma<<<dim3(HW / 16, CCH / 16, NB), 32, 0, stream>>>(
        cur, w_enc[i], z, Hin, Ho);
    gemm1x1_wmma<<<dim3(HW / 16, 1, NB), 32, 0, stream>>>(
        z, w_qh[i], zh, CCH, CCH, HW, 0);
    quantize_logits_wmma<<<dim3(HW / 16, NK / 256, NB * 2), 32, 0, stream>>>(
        zh, w_q[i], u[i], out_logit[i], packb[i], HW);
    decode_code_kernel<<<dim3(ncell / 256), 256, 0, stream>>>(
        packb[i], code_i[i], out_code[i]);
    if (i < 2) {
      gemm1x1_wmma<<<dim3(HW / 16, 1, NB), 32, 0, stream>>>(
          z, w_lh[i], lh, CCH, CCH, HW, 0);
      float* nxt = (i == 0) ? cur1 : cur2;
      gather_dq_kernel<<<dim3((NB * CCH * HW) / 256), 256, 0, stream>>>(
          lh, w_dq[i], code_i[i], nxt, HW);
      cur = nxt;
      Hin = Ho;
    }
  }

  // ================= decoder =================
  {  // step 0: w_dq0/w_dh0/w_rest0 with code2 (8x8 -> 16x16)
    const int Hq = 8, HW = 64;
    gather_dq_kernel<<<dim3((NB * CCH * HW) / 256), 256, 0, stream>>>(
        nullptr, w_dq[0], code_i[2], lh, HW);
    gemm1x1_wmma<<<dim3(HW / 16, 1, NB), 32, 0, stream>>>(
        lh, w_dh[0], zh, CCH, CCH, HW, 0);
    convT2x2_wmma<<<dim3(HW / 16, 1, NB), 32, 0, stream>>>(
        zh, w_rest[0], f16b, Hq);
  }
  {  // step 1: w_dq1/w_dh1/w_side1/w_rest1 with code1 (16x16 -> 32x32)
    const int Hq = 16, HW = 256;
    gather_dq_kernel<<<dim3((NB * CCH * HW) / 256), 256, 0, stream>>>(
        nullptr, w_dq[1], code_i[1], lh, HW);
    gemm1x1_wmma<<<dim3(HW / 16, 1, NB), 32, 0, stream>>>(
        lh, w_dh[1], zh, CCH, CCH, HW, 0);
    gemm1x1_wmma<<<dim3(HW / 16, 1, NB), 32, 0, stream>>>(
        f16b, w_side[0], zh, CCH, CCH, HW, 1);
    convT2x2_wmma<<<dim3(HW / 16, 1, NB), 32, 0, stream>>>(
        zh, w_rest[1], f32b, Hq);
  }
  {  // step 2: w_dq2/w_dh2/w_side2/w_rest2 with code0 (32x32 -> 64x64)
    const int Hq = 32, HW = 1024;
    gather_dq_kernel<<<dim3((NB * CCH * HW) / 256), 256, 0, stream>>>(
        nullptr, w_dq[2], code_i[0], lh, HW);
    gemm1x1_wmma<<<dim3(HW / 16, 1, NB), 32, 0, stream>>>(
        lh, w_dh[2], zh, CCH, CCH, HW, 0);
    gemm1x1_wmma<<<dim3(HW / 16, 1, NB), 32, 0, stream>>>(
        f32b, w_side[1], zh, CCH, CCH, HW, 1);
    convT2x2_wmma<<<dim3(HW / 16, 1, NB), 32, 0, stream>>>(
        zh, w_rest[2], out_former, Hq);
  }
}